// Tokenizer_25323127177637
// MI455X (gfx1250) — compile-verified
//
#include <hip/hip_runtime.h>

#define C_CELLS 128
#define G_GENES 2000
#define E_DIM   256
#define NBINS   20
#define HIDDEN  64
#define NB1     19   // NUM_BINS-1

typedef float v2f __attribute__((ext_vector_type(2)));
typedef float v8f __attribute__((ext_vector_type(8)));

// ---------------------------------------------------------------------------
// Kernel 1: condition rows.  out[c, 0, :] = cond_embed[cond_idx[c], :]
// ---------------------------------------------------------------------------
__global__ __launch_bounds__(E_DIM) void tok_cond_fill(
    const int* __restrict__ cond_idx,
    const float* __restrict__ cond_embed,
    float* __restrict__ out)
{
    int c = blockIdx.x;
    int e = threadIdx.x;
    out[(long)c * (G_GENES + 1) * E_DIM + e] = cond_embed[cond_idx[c] * E_DIM + e];
}

// ---------------------------------------------------------------------------
// Kernel 2: main tokenizer. One wave32 per 16-token tile.
//   per-lane: MLP(1->64)->19 logits -> softmax -> probs[20]
//   per-wave: 16 x V_WMMA_F32_16X16X4_F32 N-tiles over E, K=20 in 5 steps
// ---------------------------------------------------------------------------
__global__ __launch_bounds__(256) void tok_main(
    const float* __restrict__ expr,
    const float* __restrict__ w1,  const float* __restrict__ b1,
    const float* __restrict__ w2,  const float* __restrict__ b2,
    const float* __restrict__ gene_embed,
    const float* __restrict__ bin_embed,
    float* __restrict__ out)
{
    __shared__ float sw1[HIDDEN];
    __shared__ float sb1[HIDDEN];
    __shared__ float sw2[HIDDEN * NB1];
    __shared__ float sb2[NB1];
    __shared__ float sbin[NBINS * E_DIM];   // 20 x 256 = 20 KB

    const int tid = threadIdx.x;

    // cooperative staging of shared operands into LDS
    for (int i = tid; i < HIDDEN; i += 256) { sw1[i] = w1[i]; sb1[i] = b1[i]; }
    for (int i = tid; i < HIDDEN * NB1; i += 256) sw2[i] = w2[i];
    if (tid < NB1) sb2[tid] = b2[tid];
    for (int i = tid; i < NBINS * E_DIM; i += 256) sbin[i] = bin_embed[i];
    __syncthreads();

    const int wave = tid >> 5;
    const int lane = tid & 31;
    const int hf   = lane >> 4;     // which half-wave (K pair selector)
    const int m    = lane & 15;     // token row within tile / N column for B & D

    const int tileBase = (blockIdx.x * 8 + wave) * 16;   // first token of tile
    const int tok = tileBase + m;                        // this lane's token
    const float x = expr[tok];                           // expr is (C,G) row-major

    // ---- per-token MLP + softmax (duplicated across the two half-waves,
    //      which is exactly the A-fragment layout WMMA wants) ----
    float probs[NBINS];
    {
        float logit[NB1];
        #pragma unroll
        for (int k = 0; k < NB1; k++) logit[k] = sb2[k];
        for (int j = 0; j < HIDDEN; j++) {
            float h = fmaf(x, sw1[j], sb1[j]);
            h = h > 0.0f ? h : 0.01f * h;                // leaky_relu slope 0.01
            #pragma unroll
            for (int k = 0; k < NB1; k++)
                logit[k] = fmaf(h, sw2[j * NB1 + k], logit[k]);
        }
        float mx = logit[0];
        #pragma unroll
        for (int k = 1; k < NB1; k++) mx = fmaxf(mx, logit[k]);
        float sum = 0.0f;
        #pragma unroll
        for (int k = 0; k < NB1; k++) { logit[k] = __expf(logit[k] - mx); sum += logit[k]; }
        const float inv = 1.0f / sum;
        const bool nz = (x != 0.0f);
        probs[0] = nz ? 0.0f : 1.0f;
        #pragma unroll
        for (int k = 0; k < NB1; k++) probs[k + 1] = nz ? logit[k] * inv : 0.0f;
    }

    // ---- per-row output/gene bases for the 8 rows this lane will store ----
    int outbase[8], genebase[8];
    #pragma unroll
    for (int v = 0; v < 8; v++) {
        int t = tileBase + v + 8 * hf;        // row M = v + 8*hf of the D tile
        int c = t / G_GENES;
        int g = t - c * G_GENES;
        outbase[v]  = (c * (G_GENES + 1) + 1 + g) * E_DIM;
        genebase[v] = g * E_DIM;
    }

    // ---- WMMA: expr_emb tile = probs(16x20) @ bin_embed(20x16) per N-tile ----
    const int kbase_lds = 2 * hf;             // K offset contributed by half-wave
    #pragma unroll 1
    for (int nt = 0; nt < 16; nt++) {
        const int ncol = nt * 16 + m;         // this lane's E column
        v8f acc = {};                         // C = 0
        #pragma unroll
        for (int ks = 0; ks < 5; ks++) {
            // A fragment: lane holds probs[K], probs[K+1] with K = ks*4 + 2*hf.
            // Select with constant indices so probs[] stays in VGPRs.
            v2f a, b;
            a.x = hf ? probs[ks * 4 + 2] : probs[ks * 4 + 0];
            a.y = hf ? probs[ks * 4 + 3] : probs[ks * 4 + 1];
            // B fragment: bin_embed[K][ncol], bin_embed[K+1][ncol] from LDS.
            const int kb = ks * 4 + kbase_lds;
            b.x = sbin[kb * E_DIM + ncol];
            b.y = sbin[(kb + 1) * E_DIM + ncol];
            acc = __builtin_amdgcn_wmma_f32_16x16x4_f32(
                false, a, false, b, (short)0, acc, false, false);
        }
        // D layout: lane L -> N = L&15, VGPR v -> row M = v + 8*(L>>4)
        #pragma unroll
        for (int v = 0; v < 8; v++) {
            float val = acc[v] + gene_embed[genebase[v] + ncol];
            out[outbase[v] + ncol] = val;
        }
    }
}

// ---------------------------------------------------------------------------
extern "C" void kernel_launch(void* const* d_in, const int* in_sizes, int n_in,
                              void* d_out, int out_size, void* d_ws, size_t ws_size,
                              hipStream_t stream)
{
    const int*   cond_idx   = (const int*)  d_in[0];
    const float* expr       = (const float*)d_in[1];
    const float* w1         = (const float*)d_in[2];
    const float* b1         = (const float*)d_in[3];
    const float* w2         = (const float*)d_in[4];
    const float* b2         = (const float*)d_in[5];
    const float* gene_embed = (const float*)d_in[6];
    const float* bin_embed  = (const float*)d_in[7];
    const float* cond_embed = (const float*)d_in[8];
    float* out = (float*)d_out;

    tok_cond_fill<<<C_CELLS, E_DIM, 0, stream>>>(cond_idx, cond_embed, out);

    // 128*2000 tokens / (8 waves * 16 tokens) = 2000 blocks, exact cover
    tok_main<<<2000, 256, 0, stream>>>(expr, w1, b1, w2, b2,
                                       gene_embed, bin_embed, out);
}